// LightGCN_4518305595940
// MI455X (gfx1250) — compile-verified
//
#include <hip/hip_runtime.h>

// LightGCN on MI455X (gfx1250, wave32).
// Roofline: ~8 GB of gather/scatter traffic vs 1.9 GFLOP -> bandwidth bound.
// The 76.8 MB embedding table fits in the 192 MB L2, so COO SpMM with
// hardware GLOBAL_ATOMIC_ADD_F32 (agent scope, no-return) is the right path.
// WMMA (V_WMMA_F32_16X16X4_F32) is used where dense matrix math exists:
// the final batched 16-row dot products (diagonal of U * I^T).
//
// NOTE: k_spmm is intentionally the LAST function in this file so the
// disassembly snippet window shows its atomic lowering.

#define NUM_USERS 100000
#define NUM_ITEMS 200000
#define NNODES    (NUM_USERS + NUM_ITEMS)
#define D         64
#define BATCH     16384
// n_layers is a Python constant (3) in the reference.

typedef __attribute__((ext_vector_type(2))) float v2f;
typedef __attribute__((ext_vector_type(8))) float v8f;

// ---------------------------------------------------------------- zero buffer
__global__ void k_zero(float4* __restrict__ p, long n4) {
  long i = (long)blockIdx.x * blockDim.x + threadIdx.x;
  long stride = (long)gridDim.x * blockDim.x;
  float4 z = make_float4(0.f, 0.f, 0.f, 0.f);
  for (; i < n4; i += stride) p[i] = z;
}

// ------------------------------------------- per-sample accumulator init (=)
__global__ void k_init_acc(const int* __restrict__ users, const int* __restrict__ items,
                           const float* __restrict__ ue, const float* __restrict__ ie,
                           float* __restrict__ uacc, float* __restrict__ iacc) {
  int t = blockIdx.x * blockDim.x + threadIdx.x;
  if (t >= BATCH * D) return;
  int b = t >> 6;
  int d = t & 63;
  uacc[t] = ue[(size_t)users[b] * D + d];
  iacc[t] = ie[(size_t)items[b] * D + d];
}

// ------------------------------------------ per-sample accumulator update (+=)
__global__ void k_acc_add(const int* __restrict__ users, const int* __restrict__ items,
                          const float* __restrict__ y,
                          float* __restrict__ uacc, float* __restrict__ iacc) {
  int t = blockIdx.x * blockDim.x + threadIdx.x;
  if (t >= BATCH * D) return;
  int b = t >> 6;
  int d = t & 63;
  uacc[t] += y[(size_t)users[b] * D + d];
  iacc[t] += y[((size_t)NUM_USERS + items[b]) * D + d];
}

// ------------------------------------------------------- batched dot via WMMA
// One wave handles 16 (user,item) pairs: C = U(16x64) * I^T(64x16) via
// 16 x V_WMMA_F32_16X16X4_F32, then take the diagonal.
// A layout (16x4 f32): lane-half h supplies K=2h (comp 0) and K=2h+1 (comp 1),
// lane[3:0] = M row. B (4x16) mirrors it with lane[3:0] = N col; since
// B[k][n] = I[n][k] the addressing is identical to A with I instead of U.
// C/D layout: VGPR r, lanes 0-15 -> (M=r, N=lane); lanes 16-31 -> (M=r+8,
// N=lane-16).  Diagonal: c[r]@lane r -> gamma[r]; c[r]@lane r+24 -> gamma[r+8].
__global__ __launch_bounds__(32)
void k_dot_wmma(const float* __restrict__ uacc, const float* __restrict__ iacc,
                float* __restrict__ out, float scale) {
  int blk  = blockIdx.x;            // 16 pairs per block (one wave)
  int lane = threadIdx.x;           // 0..31
  int half = lane >> 4;             // lane-half h
  int lo   = lane & 15;             // M (for A) / N (for B)
  const float* U = uacc + (size_t)blk * 16 * D;
  const float* I = iacc + (size_t)blk * 16 * D;

  v8f c = {};
  for (int kk = 0; kk < D; kk += 4) {
    int k0 = kk + 2 * half;
    v2f a, b;
    a.x = U[(size_t)lo * D + k0];
    a.y = U[(size_t)lo * D + k0 + 1];
    b.x = I[(size_t)lo * D + k0];
    b.y = I[(size_t)lo * D + k0 + 1];
    // (neg_a, A, neg_b, B, c_mod, C, reuse_a, reuse_b)
    c = __builtin_amdgcn_wmma_f32_16x16x4_f32(false, a, false, b,
                                              (short)0, c, false, false);
  }

  float* o = out + (size_t)blk * 16;
#pragma unroll
  for (int r = 0; r < 8; ++r) {
    float v = c[r] * scale;
    if (lane == r)      o[r]     = v;   // diag M=N=r
    if (lane == r + 24) o[r + 8] = v;   // diag M=N=r+8
  }
}

// ------------------------------------------------------------------ COO SpMM
// 2 edges per wave iteration: lane-half selects the edge, lane[3:0] selects a
// float4 chunk of the 64-float row.  One b128 gather instruction serves two
// edges' rows; accumulation uses native f32 global atomics (unsafeAtomicAdd
// -> GLOBAL_ATOMIC_ADD_F32, no-return / STOREcnt).
__global__ void k_spmm(const int* __restrict__ erow, const int* __restrict__ ecol,
                       const float* __restrict__ evals,
                       const float* __restrict__ xu, const float* __restrict__ xi,
                       int split, float* __restrict__ y, int nnz) {
  int gid   = blockIdx.x * blockDim.x + threadIdx.x;
  long wave = gid >> 5;
  int lane  = threadIdx.x & 31;
  int half  = lane >> 4;            // which of the pair's 2 edges
  int q     = lane & 15;            // float4 chunk within the row
  long nw   = ((long)gridDim.x * blockDim.x) >> 5;

  for (long base = wave * 2; base < nnz; base += 2 * nw) {
    long e = base + half;
    // Software pipeline: prefetch next pair's source rows (global_prefetch_b8).
    long en = base + 2 * nw + half;
    if (en < nnz) {
      int cn = ecol[en];
      const float* pn = (cn < split) ? (xu + (size_t)cn * D)
                                     : (xi + (size_t)(cn - split) * D);
      __builtin_prefetch(pn + (size_t)q * 4, 0, 3);
    }
    if (e < nnz) {
      int   c = ecol[e];
      int   r = erow[e];
      float v = evals[e];
      const float* xb = (c < split) ? (xu + (size_t)c * D)
                                    : (xi + (size_t)(c - split) * D);
      float4 xv = ((const float4*)xb)[q];
      float* yb = y + (size_t)r * D + (size_t)q * 4;
      unsafeAtomicAdd(yb + 0, v * xv.x);
      unsafeAtomicAdd(yb + 1, v * xv.y);
      unsafeAtomicAdd(yb + 2, v * xv.z);
      unsafeAtomicAdd(yb + 3, v * xv.w);
    }
  }
}

// ---------------------------------------------------------------------- driver
extern "C" void kernel_launch(void* const* d_in, const int* in_sizes, int n_in,
                              void* d_out, int out_size, void* d_ws, size_t ws_size,
                              hipStream_t stream) {
  const int*   users = (const int*)d_in[0];
  const int*   items = (const int*)d_in[1];
  const int*   erow  = (const int*)d_in[2];
  const int*   ecol  = (const int*)d_in[3];
  const float* evals = (const float*)d_in[4];
  const float* ue    = (const float*)d_in[5];
  const float* ie    = (const float*)d_in[6];
  // d_in[7] = n_layers (device scalar) == 3 in the reference; fixed here.
  float* gamma = (float*)d_out;

  const int nnz = in_sizes[2];

  // Workspace layout: two N x D ping-pong buffers + compact sample accumulators.
  size_t rowBytes = (size_t)NNODES * D * sizeof(float);           // 76.8 MB
  float* buf0 = (float*)d_ws;
  float* buf1 = (float*)((char*)d_ws + rowBytes);
  float* uacc = (float*)((char*)d_ws + 2 * rowBytes);             // 4 MB
  float* iacc = uacc + (size_t)BATCH * D;                         // 4 MB

  const long n4        = (long)NNODES * D / 4;
  const int  ZERO_BLKS = 2048;
  const int  SPMM_BLKS = 8192;   // 64K waves, ~38 edge-pairs each
  const int  ACC_BLKS  = (BATCH * D + 255) / 256;

  // layer-0 term
  k_init_acc<<<ACC_BLKS, 256, 0, stream>>>(users, items, ue, ie, uacc, iacc);

  // layer 1: gather straight from the split embedding tables
  k_zero<<<ZERO_BLKS, 256, 0, stream>>>((float4*)buf0, n4);
  k_spmm<<<SPMM_BLKS, 256, 0, stream>>>(erow, ecol, evals, ue, ie, NUM_USERS, buf0, nnz);
  k_acc_add<<<ACC_BLKS, 256, 0, stream>>>(users, items, buf0, uacc, iacc);

  // layer 2
  k_zero<<<ZERO_BLKS, 256, 0, stream>>>((float4*)buf1, n4);
  k_spmm<<<SPMM_BLKS, 256, 0, stream>>>(erow, ecol, evals,
                                        buf0, buf0 + (size_t)NUM_USERS * D, NUM_USERS,
                                        buf1, nnz);
  k_acc_add<<<ACC_BLKS, 256, 0, stream>>>(users, items, buf1, uacc, iacc);

  // layer 3
  k_zero<<<ZERO_BLKS, 256, 0, stream>>>((float4*)buf0, n4);
  k_spmm<<<SPMM_BLKS, 256, 0, stream>>>(erow, ecol, evals,
                                        buf1, buf1 + (size_t)NUM_USERS * D, NUM_USERS,
                                        buf0, nnz);
  k_acc_add<<<ACC_BLKS, 256, 0, stream>>>(users, items, buf0, uacc, iacc);

  // gamma = dot(acc_u, acc_i) / (n_layers+1)^2 = dot / 16
  k_dot_wmma<<<BATCH / 16, 32, 0, stream>>>(uacc, iacc, gamma, 1.0f / 16.0f);
}